// SynapticPlasticity_79542794322010
// MI455X (gfx1250) — compile-verified
//
#include <hip/hip_runtime.h>
#include <hip/hip_bf16.h>
#include <math.h>

#define P_DIM 64
#define NB_DIM 256
#define N_TOT (NB_DIM * P_DIM)     // 16384
#define S_SLOTS 8192
#define TB_ROWS 128                // T*B
#define EPS 1e-6f
#define ALPHA 0.01f
#define MAX_NORM 1.0f
#define DELTA_MAX_NORM 0.1f
#define PRUNE_THR 0.01f

typedef __attribute__((ext_vector_type(2))) float v2f;
typedef __attribute__((ext_vector_type(8))) float v8f;

// ---------------------------------------------------------------------------
// Phase 1: column reductions over the (T,B)=128 rows.
// Produces modulator(raw_fb), hebbian, oja, dendritic, axonal  (each [N_TOT]).
// ---------------------------------------------------------------------------
__global__ __launch_bounds__(256) void reduce_stats_kernel(
    const float* __restrict__ vg,   // variational_gradient [128, N]
    const float* __restrict__ pg,   // post_gain            [128, N]
    const float* __restrict__ el,   // eligibility_traces   [128, N]
    const float* __restrict__ ss,   // system_states        [128, N]
    const float* __restrict__ inv,  // inverse_state_norms  [128]
    float* __restrict__ modulator,
    float* __restrict__ hebbian,
    float* __restrict__ oja,
    float* __restrict__ dendritic,
    float* __restrict__ axonal) {
  int n = blockIdx.x * blockDim.x + threadIdx.x;
  if (n >= N_TOT) return;
  float raw = 0.f, pgs = 0.f, els = 0.f, heb = 0.f, oj = 0.f;
#pragma unroll 4
  for (int r = 0; r < TB_ROWS; ++r) {
    float iv = inv[r];
    size_t off = (size_t)r * N_TOT + n;
    float v = vg[off];
    float g = pg[off];
    float e = el[off];
    float s = ss[off];
    raw += v * iv;
    pgs += g;
    els += e;
    heb += e * g * iv;
    oj  += s * iv;
  }
  modulator[n] = raw;
  hebbian[n]   = heb;
  oja[n]       = oj;
  dendritic[n] = raw * (pgs * (1.0f / TB_ROWS));
  axonal[n]    = els * (1.0f / TB_ROWS);
}

// ---------------------------------------------------------------------------
// Phase 2: trophic = |axonal @ dendritic^T| / (P+eps); EMA into new_map with
// zeroed diagonal.  f32 WMMA 16x16x4, one wave per 16x16 output tile.
//   A (16x4 f32): lane m=l%16 holds row m; VGPR0 = K = 0(+2 for hi half),
//                 VGPR1 = K = 1(+2 for hi half).
//   B (4x16 f32): lane n=l%16 holds col n; same K striping as A.
//   C/D (16x16) : VGPR v, lane l -> M = v + 8*(l>=16), N = l%16.
// ---------------------------------------------------------------------------
__global__ __launch_bounds__(32) void trophic_wmma_kernel(
    const float* __restrict__ axonal,     // [NB, P]
    const float* __restrict__ dendritic,  // [NB, P]
    const float* __restrict__ old_map,    // [NB, NB]
    float* __restrict__ new_map) {        // [NB, NB]
  const int tr   = blockIdx.y;
  const int tc   = blockIdx.x;
  const int lane = threadIdx.x;
  const int half = lane >> 4;   // 0: K+0/K+1, 1: K+2/K+3
  const int m    = lane & 15;

  const float* Arow = axonal    + (size_t)(tr * 16 + m) * P_DIM;
  const float* Brow = dendritic + (size_t)(tc * 16 + m) * P_DIM;

  v8f c = {};
#pragma unroll
  for (int kk = 0; kk < P_DIM; kk += 4) {
    const int kbase = kk + (half ? 2 : 0);
    v2f a, b;
    a.x = Arow[kbase + 0];
    a.y = Arow[kbase + 1];
    b.x = Brow[kbase + 0];
    b.y = Brow[kbase + 1];
    c = __builtin_amdgcn_wmma_f32_16x16x4_f32(
        /*neg_a=*/false, a, /*neg_b=*/false, b,
        /*c_mod=*/(short)0, c, /*reuse_a=*/false, /*reuse_b=*/false);
  }

  const float scale = 1.0f / ((float)P_DIM + EPS);
#pragma unroll
  for (int v = 0; v < 8; ++v) {
    int row = tr * 16 + v + (half ? 8 : 0);
    int col = tc * 16 + m;
    float tro = fabsf(c[v]) * scale;
    float val = old_map[(size_t)row * NB_DIM + col] * (1.0f - ALPHA) + ALPHA * tro;
    if (row == col) val = 0.0f;
    new_map[(size_t)row * NB_DIM + col] = val;
  }
}

// ---------------------------------------------------------------------------
// Block-wide sum over 256 threads (8 wave32's): shfl reduce + LDS combine.
// ---------------------------------------------------------------------------
__device__ __forceinline__ float block_reduce_sum(float v, float* red,
                                                  float* bres, int tid) {
#pragma unroll
  for (int off = 16; off > 0; off >>= 1) v += __shfl_down(v, off, 32);
  const int wid = tid >> 5;
  const int lane = tid & 31;
  if (lane == 0) red[wid] = v;
  __syncthreads();
  if (wid == 0) {
    float x = (lane < 8) ? red[lane] : 0.0f;
#pragma unroll
    for (int off = 4; off > 0; off >>= 1) x += __shfl_down(x, off, 32);
    if (lane == 0) *bres = x;
  }
  __syncthreads();
  float r = *bres;
  __syncthreads();  // red/bres are reused by the next reduction
  return r;
}

// ---------------------------------------------------------------------------
// Phase 3: per-slot weight update.  One block per slot; 256 threads, each
// owns 16 consecutive weights (fixed row i = t/4, cols jb..jb+15) so all
// global traffic is 128-bit.  Memory-bound: 268 MB total.
// ---------------------------------------------------------------------------
__global__ __launch_bounds__(256) void synapse_kernel(
    const float* __restrict__ W,           // [S, P, P]
    const int* __restrict__ rows,
    const int* __restrict__ cols,
    const unsigned char* __restrict__ active,
    const float* __restrict__ modulator,   // [NB*P]
    const float* __restrict__ hebbian,     // [NB*P]
    const float* __restrict__ oja,         // [NB*P]
    float* __restrict__ Wout) {            // [S, P, P]
  __shared__ float red[8];
  __shared__ float bres;

  const int s = blockIdx.x;
  const int t = threadIdx.x;
  const int r = rows[s];
  const int c = cols[s];
  const int i  = t >> 2;          // weight row 0..63
  const int jb = (t & 3) * 16;    // weight col base

  const float a  = modulator[r * P_DIM + i];
  const float o  = oja[r * P_DIM + i];
  const float o2 = o * o;

  const size_t wbase = (size_t)s * P_DIM * P_DIM + (size_t)i * P_DIM + jb;
  const float4* hp = (const float4*)(hebbian + c * P_DIM + jb);
  const float4* wp = (const float4*)(W + wbase);

  float h[16], w[16];
#pragma unroll
  for (int k = 0; k < 4; ++k) {
    float4 hv = hp[k];
    float4 wv = wp[k];
    h[4 * k + 0] = hv.x; h[4 * k + 1] = hv.y; h[4 * k + 2] = hv.z; h[4 * k + 3] = hv.w;
    w[4 * k + 0] = wv.x; w[4 * k + 1] = wv.y; w[4 * k + 2] = wv.z; w[4 * k + 3] = wv.w;
  }

  // Pass 1: Frobenius norm of delta.
  float dsum = 0.f;
#pragma unroll
  for (int k = 0; k < 16; ++k) {
    float d = a * h[k] - o2 * w[k];
    dsum += d * d;
  }
  const float dn2 = block_reduce_sum(dsum, red, &bres, t);
  const float scale1 = fminf(1.0f, DELTA_MAX_NORM / (sqrtf(dn2) + EPS));

  // Pass 2: apply clipped delta, prune, accumulate new norm.
  float wn[16];
  float wsum = 0.f;
#pragma unroll
  for (int k = 0; k < 16; ++k) {
    float d = a * h[k] - o2 * w[k];
    float x = w[k] + d * scale1;
    if (fabsf(x) < PRUNE_THR) x = 0.0f;
    wn[k] = x;
    wsum += x * x;
  }
  const float wn2 = block_reduce_sum(wsum, red, &bres, t);
  const float scale2 = fminf(1.0f, MAX_NORM / (sqrtf(wn2) + EPS));

  const bool act = active[s] != 0;
  float4 out4[4];
  float* outf = (float*)out4;
#pragma unroll
  for (int k = 0; k < 16; ++k) outf[k] = act ? wn[k] * scale2 : w[k];

  float4* op = (float4*)(Wout + wbase);
#pragma unroll
  for (int k = 0; k < 4; ++k) op[k] = out4[k];
}

// ---------------------------------------------------------------------------
extern "C" void kernel_launch(void* const* d_in, const int* in_sizes, int n_in,
                              void* d_out, int out_size, void* d_ws, size_t ws_size,
                              hipStream_t stream) {
  (void)in_sizes; (void)n_in; (void)out_size; (void)ws_size;

  const float* weight_values = (const float*)d_in[0];          // [S,P,P]
  const int*   weight_rows   = (const int*)d_in[1];            // [S]
  const int*   weight_cols   = (const int*)d_in[2];            // [S]
  const unsigned char* active = (const unsigned char*)d_in[3]; // [S] bool
  const float* trophic_map   = (const float*)d_in[4];          // [NB,NB]
  const float* system_states = (const float*)d_in[5];          // [128,N]
  const float* eligibility   = (const float*)d_in[6];          // [128,N]
  const float* inv_norms     = (const float*)d_in[7];          // [128]
  const float* var_grad      = (const float*)d_in[8];          // [128,N]
  const float* post_gain     = (const float*)d_in[9];          // [128,N]

  float* w_out   = (float*)d_out;                              // [S,P,P]
  float* map_out = (float*)d_out + (size_t)S_SLOTS * P_DIM * P_DIM; // [NB,NB]

  float* ws        = (float*)d_ws;       // 5 * N_TOT floats = 320 KB
  float* modulator = ws + 0 * N_TOT;
  float* hebbian   = ws + 1 * N_TOT;
  float* oja       = ws + 2 * N_TOT;
  float* dendritic = ws + 3 * N_TOT;
  float* axonal    = ws + 4 * N_TOT;

  // Phase 1: per-neuron reductions over (T,B).
  reduce_stats_kernel<<<N_TOT / 256, 256, 0, stream>>>(
      var_grad, post_gain, eligibility, system_states, inv_norms,
      modulator, hebbian, oja, dendritic, axonal);

  // Phase 2: 256x256 EMA trophic map via f32 WMMA (one wave per tile).
  trophic_wmma_kernel<<<dim3(NB_DIM / 16, NB_DIM / 16), 32, 0, stream>>>(
      axonal, dendritic, trophic_map, map_out);

  // Phase 3: per-slot synapse update (memory-bound, 268 MB traffic).
  synapse_kernel<<<S_SLOTS, 256, 0, stream>>>(
      weight_values, weight_rows, weight_cols, active,
      modulator, hebbian, oja, w_out);
}